// SpectralConv_7267084665109
// MI455X (gfx1250) — compile-verified
//
#include <hip/hip_runtime.h>
#include <math.h>

typedef float v2f __attribute__((ext_vector_type(2)));
typedef float v8f __attribute__((ext_vector_type(8)));

#define DEV static __device__ __forceinline__

constexpr int BB = 32, CIN = 128, COUT = 128, HH = 128, WW = 128;
constexpr int M1 = 32, M2 = 17;
constexpr int NMODE = M1 * M2;              // 544
constexpr int ROWS = BB * CIN * HH;         // 524288 (== BB*COUT*HH)
constexpr int NF = 48;                      // padded N for forward row-DFT (34 -> 48)
constexpr int KI = 36;                      // padded K for inverse row-DFT (34 -> 36)
constexpr float TWO_PI = 6.28318530717958647692f;

// ---- workspace layout (float offsets) ----
constexpr size_t OFF_FW   = 0;                                      // 128*48
constexpr size_t OFF_G    = OFF_FW + (size_t)WW * NF;               // 36*128
constexpr size_t OFF_WRE  = OFF_G  + (size_t)KI * WW;               // 544*128*128
constexpr size_t OFF_WIM  = OFF_WRE + (size_t)NMODE * CIN * COUT;
constexpr size_t OFF_X1   = OFF_WIM + (size_t)NMODE * CIN * COUT;   // ROWS * 34 (re/im interleaved per kw)
constexpr size_t OFF_X2RE = OFF_X1  + (size_t)ROWS * 2 * M2;        // [m][b][i]
constexpr size_t OFF_X2IM = OFF_X2RE + (size_t)NMODE * BB * CIN;
constexpr size_t OFF_YRE  = OFF_X2IM + (size_t)NMODE * BB * CIN;    // [m][b][o]
constexpr size_t OFF_YIM  = OFF_YRE + (size_t)NMODE * BB * COUT;
constexpr size_t OFF_Y1   = OFF_YIM + (size_t)NMODE * BB * COUT;    // ROWS * 36

// ---- async global->LDS path (gfx1250), with probe-safe fallback ----
#if defined(__has_builtin)
#  if __has_builtin(__builtin_amdgcn_global_load_async_to_lds_b128) && \
      __has_builtin(__builtin_amdgcn_s_wait_asynccnt)
#    define HAVE_ASYNC_LDS 1
#  endif
#endif
#ifndef HAVE_ASYNC_LDS
#  define HAVE_ASYNC_LDS 0
#endif

#if HAVE_ASYNC_LDS
// Builtin parameter type per hipcc diagnostic: pointer to int __attribute__((vector_size(16)))
typedef int b128_t __attribute__((vector_size(16)));
typedef __attribute__((address_space(1))) b128_t* gptr128;
typedef __attribute__((address_space(3))) b128_t* lptr128;
#endif

// Cooperative 16B-granular tile copy global -> LDS.
DEV void cp_tile(float* __restrict__ dst, const float* __restrict__ src, int nfloats,
                 int tid, int nthreads) {
#if HAVE_ASYNC_LDS
  int n4 = nfloats >> 2;
  for (int i = tid; i < n4; i += nthreads) {
    __builtin_amdgcn_global_load_async_to_lds_b128(
        (gptr128)(const_cast<float*>(src) + 4 * (size_t)i),
        (lptr128)(dst + 4 * (size_t)i),
        0, 0);
  }
#else
  float4* d = (float4*)dst;
  const float4* s = (const float4*)src;
  int n4 = nfloats >> 2;
  for (int i = tid; i < n4; i += nthreads) d[i] = s[i];
#endif
}

// Complete all outstanding tile copies and make them visible workgroup-wide.
DEV void cp_commit() {
#if HAVE_ASYNC_LDS
  __builtin_amdgcn_s_wait_asynccnt(0);
#endif
  __syncthreads();
}

// ---- WMMA f32 16x16x4 helpers ----
DEV v8f wmma4(v2f a, v2f b, v8f c) {
  // D = A(16x4) * B(4x16) + C ; args: (neg_a, A, neg_b, B, c_mod, C, reuse_a, reuse_b)
  return __builtin_amdgcn_wmma_f32_16x16x4_f32(false, a, false, b, (short)0, c, false, false);
}

DEV v8f vzero8() {
  v8f z;
#pragma unroll
  for (int i = 0; i < 8; ++i) z[i] = 0.f;
  return z;
}

// A fragment: 16x4 f32.  lanes0-15: M=lane, K={k0,k0+1}; lanes16-31: M=lane-16, K={k0+2,k0+3}
DEV v2f ldA(const float* smem, int ld, int rm, int k0, int lane) {
  int m = lane & 15;
  int koff = (lane & 16) ? 2 : 0;
  const float* p = smem + (rm + m) * ld + k0 + koff;
  v2f a;
  a.x = p[0];
  a.y = p[1];
  return a;
}

// B fragment: 4x16 f32.  lanes0-15: K={k0,k0+1}; lanes16-31: K={k0+2,k0+3}; N = cn + (lane&15)
DEV v2f ldB(const float* smem, int ld, int k0, int cn, int lane) {
  int n = cn + (lane & 15);
  int koff = (lane & 16) ? 2 : 0;
  v2f b;
  b.x = smem[(k0 + koff) * ld + n];
  b.y = smem[(k0 + koff + 1) * ld + n];
  return b;
}

// C/D tile: VGPR r holds rows (rm+r) [lanes0-15] and (rm+8+r) [lanes16-31], col = cn + (lane&15)
DEV void store_tile(float* base, int ld, int rm, int cn, v8f c, int lane) {
  int n = cn + (lane & 15);
  int rbase = rm + ((lane & 16) ? 8 : 0);
#pragma unroll
  for (int r = 0; r < 8; ++r) base[(size_t)(rbase + r) * ld + n] = c[r];
}

// ---------- prep: DFT matrices ----------
__global__ void k_init_dft(float* __restrict__ ws) {
  float* Fw = ws + OFF_FW;
  float* G  = ws + OFF_G;
  const float invN = 1.0f / (float)(HH * WW);  // norm='forward' scaling folded in
  for (int idx = threadIdx.x; idx < WW * NF; idx += blockDim.x) {
    int w = idx / NF, j = idx % NF;
    float v = 0.f;
    if (j < M2) {
      v = __cosf(TWO_PI * (float)(w * j) / (float)WW) * invN;
    } else if (j < 2 * M2) {
      int kw = j - M2;
      v = -__sinf(TWO_PI * (float)(w * kw) / (float)WW) * invN;
    }
    Fw[idx] = v;
  }
  for (int idx = threadIdx.x; idx < KI * WW; idx += blockDim.x) {
    int k = idx / WW, w = idx % WW;
    float v = 0.f;
    if (k < M2) {
      float c = (k == 0) ? 1.f : 2.f;            // Hermitian doubling, kw<64 so no Nyquist
      v = c * __cosf(TWO_PI * (float)(k * w) / (float)WW);
    } else if (k < 2 * M2) {
      int kw = k - M2;
      float c = (kw == 0) ? 1.f : 2.f;
      v = -c * __sinf(TWO_PI * (float)(kw * w) / (float)WW);
    }
    G[idx] = v;
  }
}

// ---------- prep: weight repack [i][o][m] -> [m][i][o] ----------
__global__ void k_repack_w(const float* __restrict__ wre_in, const float* __restrict__ wim_in,
                           float* __restrict__ ws) {
  size_t idx = (size_t)blockIdx.x * blockDim.x + threadIdx.x;
  const size_t total = (size_t)NMODE * CIN * COUT;
  if (idx >= total) return;
  int m  = (int)(idx / (CIN * COUT));
  int io = (int)(idx % (CIN * COUT));   // i*COUT + o
  size_t in_idx = (size_t)io * NMODE + m;
  ws[OFF_WRE + idx] = wre_in[in_idx];
  ws[OFF_WIM + idx] = wim_in[in_idx];
}

// ---------- stage A: row DFT (WMMA GEMM, M=ROWS,K=128,N=48pad) ----------
__global__ __launch_bounds__(128) void k_fwd_row_dft(const float* __restrict__ x,
                                                     float* __restrict__ ws) {
  __shared__ float sX[64 * WW];   // 32 KB
  __shared__ float sF[WW * NF];   // 24 KB
  const int tid = threadIdx.x, lane = tid & 31, wv = tid >> 5;
  const int rowBlock = blockIdx.x * 64;
  cp_tile(sF, ws + OFF_FW, WW * NF, tid, 128);
  cp_tile(sX, x + (size_t)rowBlock * WW, 64 * WW, tid, 128);
  cp_commit();

  const int rm = wv * 16;
  v8f acc[3];
#pragma unroll
  for (int nt = 0; nt < 3; ++nt) acc[nt] = vzero8();

#pragma unroll 4
  for (int kk = 0; kk < WW / 4; ++kk) {
    int k0 = kk * 4;
    v2f a = ldA(sX, WW, rm, k0, lane);
#pragma unroll
    for (int nt = 0; nt < 3; ++nt) {
      v2f b = ldB(sF, NF, k0, nt * 16, lane);
      acc[nt] = wmma4(a, b, acc[nt]);
    }
  }

  float* X1 = ws + OFF_X1;
  const int rloc = rm + ((lane & 16) ? 8 : 0);
#pragma unroll
  for (int nt = 0; nt < 3; ++nt) {
    int j = nt * 16 + (lane & 15);
    if (j < 2 * M2) {
      int col = (j < M2) ? (2 * j) : (2 * (j - M2) + 1);
#pragma unroll
      for (int r = 0; r < 8; ++r) {
        size_t row = (size_t)rowBlock + rloc + r;
        X1[row * (2 * M2) + col] = acc[nt][r];
      }
    }
  }
}

// ---------- stage B: column DFT keeping kh=-16..15, output [m][b][i] ----------
__global__ __launch_bounds__(M1 * M2) void k_fwd_col_dft(float* __restrict__ ws) {
  const int bi = blockIdx.x;               // b*CIN + i
  const int t = threadIdx.x;               // == mode index m = kh_idx*17 + kw
  const int kh_idx = t / M2, kw = t % M2;
  const float kh = (float)(kh_idx - M1 / 2);
  const float* X1 = ws + OFF_X1 + (size_t)bi * HH * (2 * M2) + 2 * kw;
  float sre = 0.f, sim = 0.f;
  for (int h = 0; h < HH; ++h) {
    float2 v = *(const float2*)(X1 + (size_t)h * (2 * M2));
    float s, c;
    __sincosf(-TWO_PI * (float)h * kh / (float)HH, &s, &c);
    sre += v.x * c - v.y * s;
    sim += v.x * s + v.y * c;
  }
  ws[OFF_X2RE + (size_t)t * (BB * CIN) + bi] = sre;
  ws[OFF_X2IM + (size_t)t * (BB * CIN) + bi] = sim;
}

// ---------- stage C: per-mode complex GEMM (32x128)*(128x128), WMMA ----------
__global__ __launch_bounds__(128) void k_mode_gemm(float* __restrict__ ws) {
  __shared__ float sXre[BB * CIN];    // 16 KB
  __shared__ float sXim[BB * CIN];    // 16 KB
  __shared__ float sWre[32 * COUT];   // 16 KB (K-chunk of 32)
  __shared__ float sWim[32 * COUT];   // 16 KB
  const int tid = threadIdx.x, lane = tid & 31, wv = tid >> 5;
  const int m = blockIdx.x;
  const float* wreG = ws + OFF_WRE + (size_t)m * CIN * COUT;
  const float* wimG = ws + OFF_WIM + (size_t)m * CIN * COUT;
  cp_tile(sXre, ws + OFF_X2RE + (size_t)m * BB * CIN, BB * CIN, tid, 128);
  cp_tile(sXim, ws + OFF_X2IM + (size_t)m * BB * CIN, BB * CIN, tid, 128);

  v8f cre[2][2], cim[2][2];
#pragma unroll
  for (int a = 0; a < 2; ++a)
#pragma unroll
    for (int b = 0; b < 2; ++b) { cre[a][b] = vzero8(); cim[a][b] = vzero8(); }

  const int cn0 = wv * 32;            // each wave owns 32 output cols
  for (int ch = 0; ch < CIN / 32; ++ch) {
    __syncthreads();                  // previous chunk fully consumed
    cp_tile(sWre, wreG + (size_t)ch * 32 * COUT, 32 * COUT, tid, 128);
    cp_tile(sWim, wimG + (size_t)ch * 32 * COUT, 32 * COUT, tid, 128);
    cp_commit();                      // also covers the X tiles on first iteration
#pragma unroll
    for (int kk = 0; kk < 8; ++kk) {
      int k0 = kk * 4;                // within chunk (B side)
      int kg = ch * 32 + k0;          // global K (A side)
#pragma unroll
      for (int mt = 0; mt < 2; ++mt) {
        v2f are = ldA(sXre, CIN, mt * 16, kg, lane);
        v2f aim = ldA(sXim, CIN, mt * 16, kg, lane);
        v2f aimn; aimn.x = -aim.x; aimn.y = -aim.y;   // float WMMA NEG bits must be 0
#pragma unroll
        for (int nt = 0; nt < 2; ++nt) {
          v2f bre = ldB(sWre, COUT, k0, cn0 + nt * 16, lane);
          v2f bim = ldB(sWim, COUT, k0, cn0 + nt * 16, lane);
          cre[mt][nt] = wmma4(are,  bre, cre[mt][nt]);
          cre[mt][nt] = wmma4(aimn, bim, cre[mt][nt]);
          cim[mt][nt] = wmma4(are,  bim, cim[mt][nt]);
          cim[mt][nt] = wmma4(aim,  bre, cim[mt][nt]);
        }
      }
    }
  }

  float* yre = ws + OFF_YRE + (size_t)m * BB * COUT;
  float* yim = ws + OFF_YIM + (size_t)m * BB * COUT;
#pragma unroll
  for (int mt = 0; mt < 2; ++mt)
#pragma unroll
    for (int nt = 0; nt < 2; ++nt) {
      store_tile(yre, COUT, mt * 16, cn0 + nt * 16, cre[mt][nt], lane);
      store_tile(yim, COUT, mt * 16, cn0 + nt * 16, cim[mt][nt], lane);
    }
}

// ---------- stage D: inverse column DFT -> Y1 rows [Re0..16, Im0..16, 0, 0] ----------
__global__ __launch_bounds__(HH) void k_inv_col_dft(float* __restrict__ ws) {
  __shared__ float2 sY[M1];
  const int bo = blockIdx.x;          // b*COUT + o
  const int h = threadIdx.x;
  float cs[M1], sn[M1];
#pragma unroll
  for (int k = 0; k < M1; ++k) {
    float kh = (float)(k - M1 / 2);
    __sincosf(TWO_PI * (float)h * kh / (float)HH, &sn[k], &cs[k]);
  }
  float* Y1 = ws + OFF_Y1 + ((size_t)bo * HH + h) * KI;
  for (int kw = 0; kw < M2; ++kw) {
    __syncthreads();
    if (h < M1) {
      int m = h * M2 + kw;
      float2 v;
      v.x = ws[OFF_YRE + (size_t)m * (BB * COUT) + bo];
      v.y = ws[OFF_YIM + (size_t)m * (BB * COUT) + bo];
      sY[h] = v;
    }
    __syncthreads();
    float re = 0.f, im = 0.f;
#pragma unroll
    for (int k = 0; k < M1; ++k) {
      float2 v = sY[k];
      re += v.x * cs[k] - v.y * sn[k];
      im += v.x * sn[k] + v.y * cs[k];
    }
    Y1[kw]      = re;
    Y1[M2 + kw] = im;
  }
  Y1[2 * M2]     = 0.f;   // K padding
  Y1[2 * M2 + 1] = 0.f;
}

// ---------- stage E: irfft along W (WMMA GEMM, K=36pad, N=128) + bias ----------
__global__ __launch_bounds__(128) void k_inv_row_dft(const float* __restrict__ bias,
                                                     float* __restrict__ out,
                                                     const float* __restrict__ ws) {
  __shared__ float sA[64 * KI];   // 9 KB
  __shared__ float sG[KI * WW];   // 18 KB
  const int tid = threadIdx.x, lane = tid & 31, wv = tid >> 5;
  const int rowBlock = blockIdx.x * 64;
  cp_tile(sG, ws + OFF_G, KI * WW, tid, 128);
  cp_tile(sA, ws + OFF_Y1 + (size_t)rowBlock * KI, 64 * KI, tid, 128);
  cp_commit();

  const int rm = wv * 16;
  v8f acc[8];
#pragma unroll
  for (int nt = 0; nt < 8; ++nt) acc[nt] = vzero8();

#pragma unroll
  for (int kk = 0; kk < KI / 4; ++kk) {
    int k0 = kk * 4;
    v2f a = ldA(sA, KI, rm, k0, lane);
#pragma unroll
    for (int nt = 0; nt < 8; ++nt) {
      v2f b = ldB(sG, WW, k0, nt * 16, lane);
      acc[nt] = wmma4(a, b, acc[nt]);
    }
  }

  const int o = (rowBlock / HH) % COUT;   // uniform per block (64-row tiles)
  const float bv = bias[o];
  const int rloc = rm + ((lane & 16) ? 8 : 0);
#pragma unroll
  for (int nt = 0; nt < 8; ++nt) {
    int col = nt * 16 + (lane & 15);
#pragma unroll
    for (int r = 0; r < 8; ++r) {
      size_t row = (size_t)rowBlock + rloc + r;
      out[row * WW + col] = acc[nt][r] + bv;
    }
  }
}

extern "C" void kernel_launch(void* const* d_in, const int* in_sizes, int n_in,
                              void* d_out, int out_size, void* d_ws, size_t ws_size,
                              hipStream_t stream) {
  (void)in_sizes; (void)n_in; (void)out_size; (void)ws_size;
  const float* x    = (const float*)d_in[0];
  const float* wre  = (const float*)d_in[1];
  const float* wim  = (const float*)d_in[2];
  const float* bias = (const float*)d_in[3];
  float* out = (float*)d_out;
  float* ws  = (float*)d_ws;

  hipLaunchKernelGGL(k_init_dft, dim3(1), dim3(256), 0, stream, ws);
  {
    size_t total = (size_t)NMODE * CIN * COUT;
    hipLaunchKernelGGL(k_repack_w, dim3((unsigned)((total + 255) / 256)), dim3(256), 0, stream,
                       wre, wim, ws);
  }
  hipLaunchKernelGGL(k_fwd_row_dft, dim3(ROWS / 64), dim3(128), 0, stream, x, ws);
  hipLaunchKernelGGL(k_fwd_col_dft, dim3(BB * CIN), dim3(M1 * M2), 0, stream, ws);
  hipLaunchKernelGGL(k_mode_gemm, dim3(NMODE), dim3(128), 0, stream, ws);
  hipLaunchKernelGGL(k_inv_col_dft, dim3(BB * COUT), dim3(HH), 0, stream, ws);
  hipLaunchKernelGGL(k_inv_row_dft, dim3(ROWS / 64), dim3(128), 0, stream, bias, out, ws);
}